// RNNLayer_55327768707180
// MI455X (gfx1250) — compile-verified
//
#include <hip/hip_runtime.h>

typedef __attribute__((ext_vector_type(2))) float v2f;
typedef __attribute__((ext_vector_type(8))) float v8f;

#define T_ 512
#define B_ 256
#define D_ 256
#define H_ 256

__device__ __forceinline__ float sigf(float x) {
    return 1.0f / (1.0f + __expf(-x));
}

__global__ void lstm_init_c(const float* __restrict__ c0, float* __restrict__ c_run) {
    int i = blockIdx.x * blockDim.x + threadIdx.x;
    if (i < B_ * H_) c_run[i] = c0[i];
}

// One wave (32 lanes) per 16x16 output tile of (batch, hidden).
// Computes all four gate tiles (i,f,g,o) via V_WMMA_F32_16X16X4_F32 and
// applies the fused LSTM update in-register.
__global__ __launch_bounds__(32) void lstm_step(
    const float* __restrict__ x_t,     // [B, D]   x + t*B*D
    const float* __restrict__ h_prev,  // [B, H]   feats[t-1] or h0
    const float* __restrict__ W_ih,    // [4H, D]
    const float* __restrict__ W_hh,    // [4H, H]
    const float* __restrict__ b_ih,    // [4H]
    const float* __restrict__ b_hh,    // [4H]
    const int*   __restrict__ done_t,  // [B]
    float* __restrict__ c_run,         // [B, H] in/out (cT slot of d_out)
    float* __restrict__ h_out,         // [B, H] feats[t]
    float* __restrict__ hT_out)        // [B, H] or nullptr (last step only)
{
    const int lane = threadIdx.x;      // 0..31
    const int tile = blockIdx.x;       // 0..255
    const int bm   = tile >> 4;        // batch tile index (16 rows)
    const int bn   = tile & 15;        // hidden tile index (16 cols)
    const int half = lane >> 4;        // 0: lanes 0-15, 1: lanes 16-31
    const int l16  = lane & 15;
    const int kkb  = half * 2;         // K sub-offset {0,2} per ISA A/B layout

    // A-matrix row for this lane (ISA 7.12.2: 32-bit A 16x4, M = lane%16)
    const int   arow  = bm * 16 + l16;
    const float amask = 1.0f - (float)done_t[arow];

    v8f acc[4] = {};  // gate accumulators: 0=i, 1=f, 2=g, 3=o

    // ---- gates += x_t @ W_ih^T : K over D ----
    for (int k = 0; k < D_; k += 4) {
        v2f a = *(const v2f*)(x_t + (size_t)arow * D_ + k + kkb);
#pragma unroll
        for (int g = 0; g < 4; ++g) {
            const int col = g * H_ + bn * 16 + l16;  // gate column (N)
            v2f b = *(const v2f*)(W_ih + (size_t)col * D_ + k + kkb);
            acc[g] = __builtin_amdgcn_wmma_f32_16x16x4_f32(
                false, a, false, b, (short)0, acc[g], false, false);
        }
    }

    // ---- gates += (h_prev * mask) @ W_hh^T : K over H ----
    for (int k = 0; k < H_; k += 4) {
        v2f a = *(const v2f*)(h_prev + (size_t)arow * H_ + k + kkb);
        a.x *= amask;
        a.y *= amask;
#pragma unroll
        for (int g = 0; g < 4; ++g) {
            const int col = g * H_ + bn * 16 + l16;
            v2f b = *(const v2f*)(W_hh + (size_t)col * H_ + k + kkb);
            acc[g] = __builtin_amdgcn_wmma_f32_16x16x4_f32(
                false, a, false, b, (short)0, acc[g], false, false);
        }
    }

    // ---- fused LSTM elementwise update ----
    // C/D layout (ISA 7.12.2): VGPR v, lanes 0-15 -> M=v, lanes 16-31 -> M=8+v; N=lane%16.
    const int hc = bn * 16 + l16;
    const float bi = b_ih[0 * H_ + hc] + b_hh[0 * H_ + hc];
    const float bf = b_ih[1 * H_ + hc] + b_hh[1 * H_ + hc];
    const float bg = b_ih[2 * H_ + hc] + b_hh[2 * H_ + hc];
    const float bo = b_ih[3 * H_ + hc] + b_hh[3 * H_ + hc];

#pragma unroll
    for (int v = 0; v < 8; ++v) {
        const int m    = v + half * 8;
        const int brow = bm * 16 + m;
        const float mask = 1.0f - (float)done_t[brow];

        const float iv = acc[0][v] + bi;
        const float fv = acc[1][v] + bf;
        const float gv = acc[2][v] + bg;
        const float ov = acc[3][v] + bo;

        const size_t idx = (size_t)brow * H_ + hc;
        const float cp = c_run[idx] * mask;
        const float cn = sigf(fv) * cp + sigf(iv) * tanhf(gv);
        const float hn = sigf(ov) * tanhf(cn);

        c_run[idx] = cn;
        h_out[idx] = hn;
        if (hT_out) hT_out[idx] = hn;
    }
}

extern "C" void kernel_launch(void* const* d_in, const int* in_sizes, int n_in,
                              void* d_out, int out_size, void* d_ws, size_t ws_size,
                              hipStream_t stream) {
    const float* x    = (const float*)d_in[0];  // [T,B,D]
    const float* h0   = (const float*)d_in[1];  // [1,B,H]
    const float* c0   = (const float*)d_in[2];  // [1,B,H]
    const float* W_ih = (const float*)d_in[3];  // [4H,D]
    const float* W_hh = (const float*)d_in[4];  // [4H,H]
    const float* b_ih = (const float*)d_in[5];  // [4H]
    const float* b_hh = (const float*)d_in[6];  // [4H]
    const int*   done = (const int*)d_in[7];    // [T,B]

    float* out   = (float*)d_out;
    float* feats = out;                                  // [T*B, H]
    float* hT    = out + (size_t)T_ * B_ * H_;           // [B, H]
    float* cT    = hT + (size_t)B_ * H_;                 // [B, H] doubles as running c

    // Seed running cell state from c0 (never mutate inputs).
    lstm_init_c<<<(B_ * H_ + 255) / 256, 256, 0, stream>>>(c0, cT);

    for (int t = 0; t < T_; ++t) {
        const float* hp = (t == 0) ? h0 : (feats + (size_t)(t - 1) * B_ * H_);
        lstm_step<<<256, 32, 0, stream>>>(
            x + (size_t)t * B_ * D_, hp, W_ih, W_hh, b_ih, b_hh,
            done + (size_t)t * B_, cT,
            feats + (size_t)t * B_ * H_,
            (t == T_ - 1) ? hT : nullptr);
    }
}